// GATv2_30940944401047
// MI455X (gfx1250) — compile-verified
//
#include <hip/hip_runtime.h>
#include <hip/hip_bf16.h>

typedef __attribute__((ext_vector_type(2))) float v2f;
typedef __attribute__((ext_vector_type(8))) float v8f;

#define DF 64
#define NSLOPE 0.2f

// ---------------------------------------------------------------------------
// K0: init segment-max to -inf, segment-sum to 0
// ---------------------------------------------------------------------------
__global__ void gat_init(float* __restrict__ mbuf, float* __restrict__ sbuf, int n) {
    int i = blockIdx.x * blockDim.x + threadIdx.x;
    if (i < n) {
        mbuf[i] = -__builtin_inff();
        sbuf[i] = 0.0f;
    }
}

// ---------------------------------------------------------------------------
// K1: p1 = nodes @ W1^T, p2 = nodes @ W2^T using V_WMMA_F32_16X16X4_F32.
// One block = 8 wave32s = one 16-row M-tile. Waves 0-3 -> p1 N-tiles 0..3,
// waves 4-7 -> p2 N-tiles 0..3. K=64 consumed in 16 steps of 4.
//
// A (16x4 f32, ISA layout): lanes 0-15 hold M=0..15; VGPR0/1 = K pair
//   {0,1} for lanes 0-15, {2,3} for lanes 16-31.
// B (4x16 f32): N striped across lanes (lane&15), same K-pair split.
//   B[k][n] = W_w[ncol0+n][koff+k] -> contiguous float2 per lane.
// C/D (16x16 f32, 8 VGPRs): VGPR r, lanes 0-15 -> M=r, lanes 16-31 -> M=r+8,
//   N = lane&15.
// ---------------------------------------------------------------------------
__global__ __launch_bounds__(256) void gat_proj_wmma(
    const float* __restrict__ nodes, const float* __restrict__ Ww,
    float* __restrict__ p1, float* __restrict__ p2, int nNodes) {

    const int lane  = threadIdx.x & 31;
    const int wave  = threadIdx.x >> 5;      // 0..7
    const int row0  = blockIdx.x * 16;
    const int mn    = lane & 15;             // M for A-frag, N for B-frag
    const int khalf = (lane >> 4) << 1;      // 0 or 2 (K sub-pair)
    const int isP2  = wave >> 2;
    const int ncol0 = (wave & 3) * 16;
    const int koff  = isP2 * DF;             // W1 = W_w[:, :64], W2 = W_w[:, 64:]
    float* __restrict__ outp = isP2 ? p2 : p1;

    const int arow = min(row0 + mn, nNodes - 1);
    const float* aptr = nodes + (size_t)arow * DF + khalf;
    const float* bptr = Ww + (size_t)(ncol0 + mn) * (2 * DF) + koff + khalf;

    v8f c = {};
#pragma unroll
    for (int k0 = 0; k0 < DF; k0 += 4) {
        v2f a = *(const v2f*)(aptr + k0);
        v2f b = *(const v2f*)(bptr + k0);
        // 8 args: (neg_a, A, neg_b, B, c_mod, C, reuse_a, reuse_b)
        c = __builtin_amdgcn_wmma_f32_16x16x4_f32(false, a, false, b,
                                                  (short)0, c, false, false);
    }

    const int mbase = (lane >> 4) << 3;      // 0 or 8
#pragma unroll
    for (int r = 0; r < 8; ++r) {
        const int row = row0 + mbase + r;
        if (row < nNodes)
            outp[(size_t)row * DF + ncol0 + mn] = c[r];
    }
}

// ---------------------------------------------------------------------------
// float atomic max via int punning (monotone bit encodings)
// ---------------------------------------------------------------------------
__device__ __forceinline__ void atomicMaxF(float* addr, float v) {
    if (v >= 0.0f)
        atomicMax((int*)addr, __float_as_int(v));
    else
        atomicMin((unsigned int*)addr, __float_as_uint(v));
}

// ---------------------------------------------------------------------------
// K2: per-edge score. One wave32 per edge; each lane covers 2 of 64 features.
// score = sum_k lrelu(p1[src][k] + p2[dst][k] + Wb[k]) * aw[k]
// p1/p2 gathers are L2-resident (51 MB << 192 MB L2).
// ---------------------------------------------------------------------------
__global__ __launch_bounds__(256) void gat_edge_score(
    const float* __restrict__ p1, const float* __restrict__ p2,
    const int* __restrict__ src, const int* __restrict__ dst,
    const float* __restrict__ Wb, const float* __restrict__ aw,
    float* __restrict__ scores, float* __restrict__ mbuf, int nEdges) {

    const int e = blockIdx.x * (blockDim.x >> 5) + (threadIdx.x >> 5);
    if (e >= nEdges) return;
    const int lane = threadIdx.x & 31;

    const int s = src[e];
    const int d = dst[e];

    v2f x1 = *(const v2f*)(p1 + (size_t)s * DF + lane * 2);
    v2f x2 = *(const v2f*)(p2 + (size_t)d * DF + lane * 2);
    v2f wb = *(const v2f*)(Wb + lane * 2);
    v2f av = *(const v2f*)(aw + lane * 2);

    float h0 = x1.x + x2.x + wb.x;
    float h1 = x1.y + x2.y + wb.y;
    h0 = (h0 >= 0.0f) ? h0 : NSLOPE * h0;
    h1 = (h1 >= 0.0f) ? h1 : NSLOPE * h1;
    float part = h0 * av.x + h1 * av.y;

#pragma unroll
    for (int off = 16; off > 0; off >>= 1)
        part += __shfl_xor(part, off, 32);

    if (lane == 0) {
        scores[e] = part;
        atomicMaxF(&mbuf[s], part);
    }
}

// ---------------------------------------------------------------------------
// K3: e = exp(score - m[src]); out[e] = e; s[src] += e
// ---------------------------------------------------------------------------
__global__ void gat_edge_exp(const float* __restrict__ scores,
                             const int* __restrict__ src,
                             const float* __restrict__ mbuf,
                             float* __restrict__ sbuf,
                             float* __restrict__ out, int nEdges) {
    int e = blockIdx.x * blockDim.x + threadIdx.x;
    if (e >= nEdges) return;
    const int s = src[e];
    float v = expf(scores[e] - mbuf[s]);
    out[e] = v;
    atomicAdd(&sbuf[s], v);
}

// ---------------------------------------------------------------------------
// K4: out[e] /= s[src[e]]
// ---------------------------------------------------------------------------
__global__ void gat_edge_div(const int* __restrict__ src,
                             const float* __restrict__ sbuf,
                             float* __restrict__ out, int nEdges) {
    int e = blockIdx.x * blockDim.x + threadIdx.x;
    if (e >= nEdges) return;
    out[e] = out[e] / sbuf[src[e]];
}

// ---------------------------------------------------------------------------
// Launch: inputs are {nodes, src, dst, W_w, W_b, a_w}; output = attn (f32).
// Workspace layout (floats): p1[N*64] | p2[N*64] | scores[E] | m[N] | s[N]
//  = 100000*64*2 + 1.6M + 2*100000 = ~14.6M floats = ~58.4 MB.
// ---------------------------------------------------------------------------
extern "C" void kernel_launch(void* const* d_in, const int* in_sizes, int n_in,
                              void* d_out, int out_size, void* d_ws, size_t ws_size,
                              hipStream_t stream) {
    const float* nodes = (const float*)d_in[0];
    const int*   src   = (const int*)d_in[1];
    const int*   dst   = (const int*)d_in[2];
    const float* Ww    = (const float*)d_in[3];
    const float* Wb    = (const float*)d_in[4];
    const float* aw    = (const float*)d_in[5];
    float*       out   = (float*)d_out;

    const int nNodes = in_sizes[0] / DF;
    const int nEdges = in_sizes[1];

    float* p1     = (float*)d_ws;
    float* p2     = p1 + (size_t)nNodes * DF;
    float* scores = p2 + (size_t)nNodes * DF;
    float* mbuf   = scores + (size_t)nEdges;
    float* sbuf   = mbuf + (size_t)nNodes;

    gat_init<<<(nNodes + 255) / 256, 256, 0, stream>>>(mbuf, sbuf, nNodes);

    gat_proj_wmma<<<(nNodes + 15) / 16, 256, 0, stream>>>(nodes, Ww, p1, p2, nNodes);

    // 8 edges per 256-thread block (one wave32 per edge)
    gat_edge_score<<<(nEdges + 7) / 8, 256, 0, stream>>>(
        p1, p2, src, dst, Wb, aw, scores, mbuf, nEdges);

    gat_edge_exp<<<(nEdges + 255) / 256, 256, 0, stream>>>(
        scores, src, mbuf, sbuf, out, nEdges);

    gat_edge_div<<<(nEdges + 255) / 256, 256, 0, stream>>>(
        src, sbuf, out, nEdges);
}